// Custom_CE_Loss_60413009985935
// MI455X (gfx1250) — compile-verified
//
#include <hip/hip_runtime.h>
#include <hip/hip_bf16.h>
#include <math.h>

typedef float v2f __attribute__((ext_vector_type(2)));
typedef float v8f __attribute__((ext_vector_type(8)));

// ---------------- workspace layout ----------------
// [0, 64)                : double acc[8]  (0:ce_sum 1:S1_txf 2:S2_txf 3:S1_img 4:S2_img)
// [64, 64+C*D*4)         : class_sums -> protos (in place)
// then C floats          : class_counts
// then C floats          : txf_norms
// then C floats          : proto_norms

__global__ __launch_bounds__(256) void zero_ws_kernel(float* buf, int nf, double* acc) {
    int idx = blockIdx.x * 256 + threadIdx.x;
    if (idx < nf) buf[idx] = 0.0f;
    if (idx < 8)  acc[idx] = 0.0;
}

// ---------------- cross entropy (online softmax, one block per row) --------
__global__ __launch_bounds__(256) void ce_kernel(const float* __restrict__ logits,
                                                 const int* __restrict__ gt,
                                                 int C, double* __restrict__ acc) {
    int row = blockIdx.x;
    const float* lr = logits + (size_t)row * C;
    int tid = threadIdx.x;

    float m = -INFINITY, s = 0.0f;
    for (int c = tid; c < C; c += 256) {
        float x  = lr[c];
        float nm = fmaxf(m, x);
        s = s * __expf(m - nm) + __expf(x - nm);
        m = nm;
    }
    __shared__ float sm[256], ss[256];
    sm[tid] = m; ss[tid] = s;
    __syncthreads();
    for (int o = 128; o > 0; o >>= 1) {
        if (tid < o) {
            float m2 = sm[tid + o], s2 = ss[tid + o];
            float M  = fmaxf(sm[tid], m2);
            ss[tid]  = ss[tid] * __expf(sm[tid] - M) + s2 * __expf(m2 - M);
            sm[tid]  = M;
        }
        __syncthreads();
    }
    if (tid == 0) {
        float x  = lr[gt[row]];
        float lp = x - sm[0] - __logf(ss[0]);
        atomicAdd(acc, (double)(-lp));
    }
}

// ---------------- segment sum: class sums + counts -------------------------
__global__ __launch_bounds__(256) void segsum_kernel(const float* __restrict__ imf,
                                                     const int* __restrict__ gt,
                                                     int N, int D,
                                                     float* __restrict__ sums,
                                                     float* __restrict__ counts) {
    int idx = blockIdx.x * 256 + threadIdx.x;
    if (idx >= N * D) return;
    int i = idx / D;
    int d = idx - i * D;
    int c = gt[i];
    atomicAdd(&sums[c * D + d], imf[idx]);
    if (d == 0) atomicAdd(&counts[c], 1.0f);
}

// ---------------- protos = sums / counts (in place) ------------------------
__global__ __launch_bounds__(256) void protos_kernel(float* __restrict__ sums,
                                                     const float* __restrict__ counts,
                                                     int C, int D) {
    int idx = blockIdx.x * 256 + threadIdx.x;
    if (idx >= C * D) return;
    float cnt = counts[idx / D];
    sums[idx] = sums[idx] / fmaxf(cnt, 1.0f);
}

// ---------------- row squared norms for txf (blocks [0,C)) and protos ------
__global__ __launch_bounds__(256) void norms_kernel(const float* __restrict__ txf,
                                                    const float* __restrict__ protos,
                                                    int C, int D,
                                                    float* __restrict__ txf_norms,
                                                    float* __restrict__ proto_norms) {
    int b = blockIdx.x;
    const float* rowp;
    float* outp;
    if (b < C) { rowp = txf    + (size_t)b * D;       outp = &txf_norms[b]; }
    else       { rowp = protos + (size_t)(b - C) * D; outp = &proto_norms[b - C]; }
    float s = 0.0f;
    for (int d = threadIdx.x; d < D; d += 256) { float x = rowp[d]; s += x * x; }
    __shared__ float sh[256];
    sh[threadIdx.x] = s;
    __syncthreads();
    for (int o = 128; o > 0; o >>= 1) {
        if (threadIdx.x < o) sh[threadIdx.x] += sh[threadIdx.x + o];
        __syncthreads();
    }
    if (threadIdx.x == 0) *outp = sh[0];
}

// ---------------- WMMA Gram + strict-upper distance moments ----------------
// One wave per 16x16 output tile; only tj >= ti tile pairs do work.
// Gram tile = X[ti*16..][:] . X[tj*16..][:]^T  via V_WMMA_F32_16X16X4_F32.
// A 16x4 layout : lane l holds A[l%16][2*(l/16)], A[l%16][2*(l/16)+1]
// B 4x16 layout : lane l holds B[2*(l/16)][l%16], B[2*(l/16)+1][l%16]
// C/D 16x16     : VGPR r, lane l -> row = r + 8*(l/16), col = l%16
__global__ __launch_bounds__(32) void gram_dist_kernel(const float* __restrict__ X,
                                                       const float* __restrict__ norms,
                                                       int n, int D,
                                                       double* __restrict__ s1_acc,
                                                       double* __restrict__ s2_acc) {
    int ti = blockIdx.y, tj = blockIdx.x;
    if (tj < ti) return;

    int lane = threadIdx.x;        // 0..31
    int m15  = lane & 15;
    int koff = (lane >> 4) << 1;   // 0 or 2

    int arow = ti * 16 + m15; if (arow > n - 1) arow = n - 1;   // clamp pads (masked later)
    int brow = tj * 16 + m15; if (brow > n - 1) brow = n - 1;

    const float* ap = X + (size_t)arow * D + koff;
    const float* bp = X + (size_t)brow * D + koff;

    v8f c = {0.f, 0.f, 0.f, 0.f, 0.f, 0.f, 0.f, 0.f};
#pragma unroll 4
    for (int k = 0; k < D; k += 4) {
        v2f a = *(const v2f*)(ap + k);
        v2f b = *(const v2f*)(bp + k);
        c = __builtin_amdgcn_wmma_f32_16x16x4_f32(false, a, false, b,
                                                  (short)0, c, false, false);
    }

    // strict-upper masked accumulation of d and d^2
    int colg = tj * 16 + m15;
    float ncol = (colg < n) ? norms[colg] : 0.0f;
    float s1 = 0.0f, s2 = 0.0f;
#pragma unroll
    for (int r = 0; r < 8; ++r) {
        int rowg = ti * 16 + r + ((lane >> 4) << 3);
        if (colg > rowg && colg < n && rowg < n) {
            float dd = norms[rowg] + ncol - 2.0f * c[r];
            s1 += dd;
            s2 += dd * dd;
        }
    }
    // wave32 reduction
    for (int o = 16; o > 0; o >>= 1) {
        s1 += __shfl_xor(s1, o, 32);
        s2 += __shfl_xor(s2, o, 32);
    }
    if (lane == 0) {
        atomicAdd(s1_acc, (double)s1);
        atomicAdd(s2_acc, (double)s2);
    }
}

// ---------------- combine ---------------------------------------------------
__global__ void finalize_kernel(const double* __restrict__ acc, int N, int n,
                                float* __restrict__ out) {
    if (threadIdx.x == 0 && blockIdx.x == 0) {
        double denom = 2.0 / ((double)n * (double)n - (double)n);
        double ce  = acc[0] / (double)N;
        double mu  = denom * acc[1];
        double rw1 = denom * acc[2] - mu * mu;
        double mu1 = denom * acc[3];
        double rw2 = denom * acc[4] - 2.0 * mu * mu1 + mu * mu;
        out[0] = (float)(ce + rw1 + rw2);
    }
}

extern "C" void kernel_launch(void* const* d_in, const int* in_sizes, int n_in,
                              void* d_out, int out_size, void* d_ws, size_t ws_size,
                              hipStream_t stream) {
    const float* logits = (const float*)d_in[0];   // [N, C]
    const int*   gt     = (const int*)d_in[1];     // [N]
    const float* txf    = (const float*)d_in[2];   // [C, D]
    const float* imf    = (const float*)d_in[3];   // [N, D]
    float* out = (float*)d_out;

    const int N = in_sizes[1];
    const int C = in_sizes[0] / N;
    const int D = in_sizes[2] / C;

    double* acc        = (double*)d_ws;
    float*  class_sums = (float*)((char*)d_ws + 64);        // C*D floats (-> protos)
    float*  counts     = class_sums + (size_t)C * D;        // C floats
    float*  txf_norms  = counts + C;                        // C floats
    float*  prot_norms = txf_norms + C;                     // C floats

    // 1. zero class_sums + counts + accumulators
    int nzero = C * D + C;
    zero_ws_kernel<<<(nzero + 255) / 256, 256, 0, stream>>>(class_sums, nzero, acc);

    // 2. cross-entropy sum -> acc[0]
    ce_kernel<<<N, 256, 0, stream>>>(logits, gt, C, &acc[0]);

    // 3. segment sums + counts
    int nseg = N * D;
    segsum_kernel<<<(nseg + 255) / 256, 256, 0, stream>>>(imf, gt, N, D, class_sums, counts);

    // 4. prototypes (in place)
    protos_kernel<<<(C * D + 255) / 256, 256, 0, stream>>>(class_sums, counts, C, D);

    // 5. row norms for txf and protos
    norms_kernel<<<2 * C, 256, 0, stream>>>(txf, class_sums, C, D, txf_norms, prot_norms);

    // 6+7. WMMA Gram -> strict-upper distance moments
    int T = (C + 15) / 16;
    dim3 tgrid(T, T);
    gram_dist_kernel<<<tgrid, 32, 0, stream>>>(txf,        txf_norms,  C, D, &acc[1], &acc[2]);
    gram_dist_kernel<<<tgrid, 32, 0, stream>>>(class_sums, prot_norms, C, D, &acc[3], &acc[4]);

    // 8. combine
    finalize_kernel<<<1, 1, 0, stream>>>(acc, N, C, out);
}